// MotionNetLoss_20375324852449
// MI455X (gfx1250) — compile-verified
//
#include <hip/hip_runtime.h>
#include <hip/hip_bf16.h>

typedef __attribute__((ext_vector_type(2))) float v2f;
typedef __attribute__((ext_vector_type(8))) float v8f;

#define RES_FLOATS (16*33*128*128)          /* 8,650,752 resized tensor  */
#define STATS_OFF  RES_FLOATS               /* 480*6 channel statistics  */
#define SMOOTH_OFF (STATS_OFF + 480*6)      /* 160 smoothness partials   */

// Deterministic 256->1 fp32 reduction.  B-matrix layout for
// V_WMMA_F32_16X16X4_F32 (wave32): VGPR0 = rows 0,1 (lanes 0-15 / 16-31),
// VGPR1 = rows 2,3.  With A = ones(16x4), D[m][n] = colsum_n(B); chaining C
// accumulates 4 chunks of 64 -> column n holds sum of lds[m*16+n] over m.
// Row 0 of D (acc[0], lanes 0-15) then needs a 16-lane fold.
__device__ __forceinline__ float wave_reduce_256_wmma(const float* lds, int lane) {
  v2f ones; ones.x = 1.0f; ones.y = 1.0f;
  v8f acc = {0.0f, 0.0f, 0.0f, 0.0f, 0.0f, 0.0f, 0.0f, 0.0f};
#pragma unroll
  for (int ch = 0; ch < 4; ++ch) {
    v2f bm;
    bm.x = lds[ch * 64 + lane];        // B rows 0,1 of this chunk
    bm.y = lds[ch * 64 + 32 + lane];   // B rows 2,3 of this chunk
    acc = __builtin_amdgcn_wmma_f32_16x16x4_f32(
        false, ones, false, bm, (short)0, acc, false, false);
  }
  float v = acc[0];                     // lanes 0-15: 16 column partials
  v += __shfl_xor(v, 8, 32);
  v += __shfl_xor(v, 4, 32);
  v += __shfl_xor(v, 2, 32);
  v += __shfl_xor(v, 1, 32);
  return v;                             // lane 0 holds the total
}

__device__ __forceinline__ float charb(float g) {
  return powf(g * g + 1.0e-6f, 0.4f);  // (g^2 + EPS^2)^ALPHA, EPS=1e-3, ALPHA=0.4
}

// ---------------- Kernel A: bilinear resize 256x256 -> 128x128 (align corners)
__global__ void resize_kernel(const float* __restrict__ img, float* __restrict__ res) {
  int idx = blockIdx.x * 256 + threadIdx.x;      // < 16*33*128*128
  int j  = idx & 127;
  int i  = (idx >> 7) & 127;
  int bc = idx >> 14;                            // 0..527
  const float S = 255.0f / 127.0f;
  float fy = (float)i * S, fx = (float)j * S;
  int y0 = (int)fy; if (y0 > 254) y0 = 254;
  int x0 = (int)fx; if (x0 > 254) x0 = 254;
  float wy = fy - (float)y0, wx = fx - (float)x0;
  const float* s = img + (size_t)bc * 65536 + (size_t)y0 * 256 + x0;
  float v00 = s[0], v01 = s[1], v10 = s[256], v11 = s[257];
  float c0 = v00 * (1.0f - wy) + v10 * wy;       // blend rows first (as reference)
  float c1 = v01 * (1.0f - wy) + v11 * wy;
  res[idx] = c0 * (1.0f - wx) + c1 * wx;
}

// ---------------- Kernel B: warp + SSIM moments + pixel loss, one block per (b,c)
__global__ void warp_stats_kernel(const float* __restrict__ res,
                                  const float* __restrict__ flows,
                                  float* __restrict__ stats) {
  __shared__ float red[6][256];
  int tid = threadIdx.x;
  int blk = blockIdx.x;                 // 0..479
  int b = blk / 30, c = blk % 30, f = c / 3;
  const float* refc = res + (size_t)(b * 33 + c) * 16384;
  const float* srcc = res + (size_t)(b * 33 + 3 + c) * 16384;
  const float* flx  = flows + (size_t)(b * 20 + 2 * f) * 16384;
  const float* fly  = flx + 16384;
  float sx = 0, sy = 0, sxx = 0, syy = 0, sxy = 0, sp = 0;
#pragma unroll 4
  for (int k = 0; k < 64; ++k) {
    int p = tid + k * 256;
    int i = p >> 7, j = p & 127;
    float x  = refc[p];
    float gx = (float)j + flx[p];
    float gy = (float)i + fly[p];
    float x0f = floorf(gx), y0f = floorf(gy);
    float wx = gx - x0f, wy = gy - y0f;          // weights from unclipped floor
    int x0 = (int)x0f; x0 = x0 < 0 ? 0 : (x0 > 127 ? 127 : x0);
    int y0 = (int)y0f; y0 = y0 < 0 ? 0 : (y0 > 127 ? 127 : y0);
    int x1 = x0 + 1; if (x1 > 127) x1 = 127;
    int y1 = y0 + 1; if (y1 > 127) y1 = 127;
    const float* r0 = srcc + y0 * 128;
    const float* r1 = srcc + y1 * 128;
    float v00 = r0[x0], v01 = r0[x1], v10 = r1[x0], v11 = r1[x1];
    float top = v00 * (1.0f - wx) + v01 * wx;
    float bot = v10 * (1.0f - wx) + v11 * wx;
    float y   = top * (1.0f - wy) + bot * wy;
    sx += x; sy += y; sxx += x * x; syy += y * y; sxy += x * y;
    float d = x - y;
    sp += powf(d * d + 1.0e-6f, 0.4f);
  }
  red[0][tid] = sx;  red[1][tid] = sy;  red[2][tid] = sxx;
  red[3][tid] = syy; red[4][tid] = sxy; red[5][tid] = sp;
  __syncthreads();
  if (tid < 32) {                       // full wave -> EXEC all ones for WMMA
    for (int s = 0; s < 6; ++s) {
      float t = wave_reduce_256_wmma(&red[s][0], tid);
      if (tid == 0) stats[blk * 6 + s] = t;
    }
  }
}

// ---------------- Kernel C: smoothness (gradient along flow-index axis and H)
__global__ void smooth_kernel(const float* __restrict__ flows, float* __restrict__ part) {
  __shared__ float red[256];
  int tid = threadIdx.x;
  int g = blockIdx.x;                   // 0..159
  int b = g / 10, f = g % 10;
  const float* base = flows + (size_t)b * 20 * 16384;
  int fm = f > 0 ? f - 1 : 0;
  int fp = f < 9 ? f + 1 : 9;
  float s1 = (f == 0 || f == 9) ? 1.0f : 0.5f;
  const float* cx  = base + (size_t)(2 * f)  * 16384;  // flow_x channel f
  const float* cy  = cx + 16384;                       // flow_y channel f
  const float* cxm = base + (size_t)(2 * fm) * 16384;
  const float* cxp = base + (size_t)(2 * fp) * 16384;
  const float* cym = cxm + 16384;
  const float* cyp = cxp + 16384;
  float acc = 0.0f;
  for (int k = 0; k < 64; ++k) {
    int p = tid + k * 256;
    int i = p >> 7, j = p & 127;
    int im = i > 0 ? i - 1 : 0;
    int ip = i < 127 ? i + 1 : 127;
    float s2 = (i == 0 || i == 127) ? 1.0f : 0.5f;
    float g1x = s1 * (cxp[p] - cxm[p]);
    float g1y = s1 * (cyp[p] - cym[p]);
    float g2x = s2 * (cx[ip * 128 + j] - cx[im * 128 + j]);
    float g2y = s2 * (cy[ip * 128 + j] - cy[im * 128 + j]);
    acc += charb(g1x) + charb(g2x) + charb(g1y) + charb(g2y);
  }
  red[tid] = acc;
  __syncthreads();
  if (tid < 32) {
    float t = wave_reduce_256_wmma(red, tid);
    if (tid == 0) part[g] = t;
  }
}

// ---------------- Kernel D: SSIM from moments + combine all partials
__global__ void finalize_kernel(const float* __restrict__ stats,
                                const float* __restrict__ smoothp,
                                float* __restrict__ out) {
  __shared__ float red[3][256];
  int tid = threadIdx.x;
  const float N = 16384.0f;
  const float C1c = 1.0e-4f, C2c = 1.0e-3f;
  float ssim_acc = 0.0f, pix_acc = 0.0f, sm_acc = 0.0f;
  for (int ch = tid; ch < 480; ch += 256) {
    const float* s = stats + ch * 6;
    float sx = s[0], sy = s[1], sxx = s[2], syy = s[3], sxy = s[4], sp = s[5];
    float mu1 = sx / N, mu2 = sy / N;
    float var1 = (sxx - sx * sx / N) / (N - 1.0f);
    float var2 = (syy - sy * sy / N) / (N - 1.0f);
    float s12  = (sxy - sx * sy / N) / N;
    float num = (2.0f * mu1 * mu2 + C1c) * (2.0f * s12 + C2c);
    float den = (mu1 * mu1 + mu2 * mu2 + C1c) * (var1 + var2 + C2c);
    ssim_acc += num / den;
    pix_acc  += sp;
  }
  for (int q = tid; q < 160; q += 256) sm_acc += smoothp[q];
  red[0][tid] = ssim_acc; red[1][tid] = pix_acc; red[2][tid] = sm_acc;
  __syncthreads();
  if (tid < 32) {
    float tot[3];
    for (int s = 0; s < 3; ++s) tot[s] = wave_reduce_256_wmma(&red[s][0], tid);
    if (tid == 0) {
      float similarity = tot[0] / 480.0f;
      float pixel      = tot[1] / (480.0f * 16384.0f);
      float smooth     = tot[2] / 2621440.0f;
      out[0] = pixel + 0.01f * smooth + similarity;   // weights (1.0, 0.01, 1.0)
    }
  }
}

extern "C" void kernel_launch(void* const* d_in, const int* in_sizes, int n_in,
                              void* d_out, int out_size, void* d_ws, size_t ws_size,
                              hipStream_t stream) {
  (void)in_sizes; (void)n_in; (void)out_size; (void)ws_size;
  const float* images = (const float*)d_in[0];   // (16,33,256,256) fp32
  const float* flows  = (const float*)d_in[1];   // (16,20,128,128) fp32
  float* ws      = (float*)d_ws;
  float* res     = ws;                 // 8,650,752 floats (34.6 MB, L2-resident)
  float* stats   = ws + STATS_OFF;     // 480*6 floats
  float* smoothp = ws + SMOOTH_OFF;    // 160 floats

  resize_kernel    <<<RES_FLOATS / 256, 256, 0, stream>>>(images, res);
  warp_stats_kernel<<<480,              256, 0, stream>>>(res, flows, stats);
  smooth_kernel    <<<160,              256, 0, stream>>>(flows, smoothp);
  finalize_kernel  <<<1,                256, 0, stream>>>(stats, smoothp, (float*)d_out);
}